// GraphCON_GCN_6253472383694
// MI455X (gfx1250) — compile-verified
//
#include <hip/hip_runtime.h>
#include <hip/hip_bf16.h>

#define NHID 128
#define DT 1.0f
#define ALPHA 1.0f
#define GAMMA 1.0f

typedef float v2f __attribute__((ext_vector_type(2)));
typedef float v8f __attribute__((ext_vector_type(8)));

__device__ __forceinline__ v8f wmma_f32_4(v2f a, v2f b, v8f c) {
    // D = A(16x4 f32) * B(4x16 f32) + C(16x16 f32), full fp32 WMMA
    return __builtin_amdgcn_wmma_f32_16x16x4_f32(false, a, false, b, (short)0, c, false, false);
}

// ---------------- pointwise kernels ----------------

__global__ __launch_bounds__(256) void init_kernel(const float* __restrict__ x,
                                                   float* __restrict__ X,
                                                   float* __restrict__ Y,
                                                   long long n) {
    long long i = (long long)blockIdx.x * 256 + threadIdx.x;
    if (i >= n) return;
    float v = tanhf(x[i]);
    X[i] = v;
    Y[i] = v;
}

__global__ __launch_bounds__(256) void fill_zero_kernel(float* __restrict__ p, long long n) {
    long long i = (long long)blockIdx.x * 256 + threadIdx.x;
    if (i < n) p[i] = 0.0f;
}

__global__ __launch_bounds__(256) void deg_count_kernel(const int* __restrict__ dst,
                                                        float* __restrict__ deg, int E) {
    int e = blockIdx.x * 256 + threadIdx.x;
    if (e >= E) return;
    atomicAdd(&deg[dst[e]], 1.0f);
}

__global__ __launch_bounds__(256) void dinv_kernel(float* __restrict__ deg_to_dinv, int N) {
    int i = blockIdx.x * 256 + threadIdx.x;
    if (i >= N) return;
    deg_to_dinv[i] = rsqrtf(deg_to_dinv[i] + 1.0f);  // self-loop: deg+1
}

// ---------------- edge scatter: agg[dst] += h[src] * dinv[src]*dinv[dst] ----------------
// One thread per (edge, 4-float chunk): 32 threads cover a 128-wide row.

__global__ __launch_bounds__(256) void edge_kernel(const int* __restrict__ src,
                                                   const int* __restrict__ dst,
                                                   const float* __restrict__ h,
                                                   const float* __restrict__ dinv,
                                                   float* __restrict__ agg, int E) {
    long long t = (long long)blockIdx.x * 256 + threadIdx.x;
    long long e = t >> 5;
    if (e >= E) return;
    int c = ((int)t & 31) * 4;
    int s = src[e];
    int d = dst[e];
    float norm = dinv[s] * dinv[d];
    const float4 hv = *(const float4*)(h + (long long)s * NHID + c);
    float* ap = agg + (long long)d * NHID + c;
    atomicAdd(ap + 0, hv.x * norm);
    atomicAdd(ap + 1, hv.y * norm);
    atomicAdd(ap + 2, hv.z * norm);
    atomicAdd(ap + 3, hv.w * norm);
}

// ---------------- WMMA GEMM: C[N x 128] = A[N x 128] * B[128 x 128] ----------------
// 256 threads = 8 waves per block; each wave computes a 16-row panel.
// B staged in LDS (64 KB); A panel held in 64 VGPRs per lane.

__global__ __launch_bounds__(256) void gemm_h_kernel(const float* __restrict__ A,
                                                     const float* __restrict__ Bg,
                                                     float* __restrict__ C, int N) {
    __shared__ float sB[NHID * NHID];
    for (int i = threadIdx.x; i < NHID * NHID; i += 256) sB[i] = Bg[i];
    __syncthreads();

    const int wave = threadIdx.x >> 5;
    const int lane = threadIdx.x & 31;
    const int m0 = (blockIdx.x * 8 + wave) * 16;
    if (m0 >= N) return;  // wave-uniform: EXEC stays all-ones for WMMA

    const int row  = m0 + (lane & 15);
    const int rrow = (row < N) ? row : (N - 1);   // clamp, keep EXEC full
    const int koff = (lane >> 4) << 1;            // 0 for lanes 0-15, 2 for 16-31
    const int col  = lane & 15;

    v2f a[32];
#pragma unroll
    for (int kb = 0; kb < 32; ++kb) {
        const float* ap = A + (long long)rrow * NHID + kb * 4 + koff;
        a[kb].x = ap[0];
        a[kb].y = ap[1];
    }

    const int rbase = m0 + ((lane >> 4) << 3);
#pragma unroll
    for (int nt = 0; nt < 8; ++nt) {
        const int n0 = nt * 16;
        v8f acc = {0.f, 0.f, 0.f, 0.f, 0.f, 0.f, 0.f, 0.f};
#pragma unroll
        for (int kb = 0; kb < 32; ++kb) {
            const int k = kb * 4 + koff;
            v2f b;
            b.x = sB[k * NHID + n0 + col];
            b.y = sB[(k + 1) * NHID + n0 + col];
            acc = wmma_f32_4(a[kb], b, acc);
        }
#pragma unroll
        for (int j = 0; j < 8; ++j) {
            const int r = rbase + j;
            if (r < N) C[(long long)r * NHID + n0 + col] = acc[j];
        }
    }
}

// ---------------- fused residual GEMM + GraphCON update ----------------
// res = h @ Wres; z = agg + h*dinv^2 + b - res - bres
// Ynew = Y + DT*(tanh(z) - ALPHA*Y - GAMMA*X); Xnew = X + DT*Ynew

__global__ __launch_bounds__(256) void combine_kernel(const float* __restrict__ h,
                                                      const float* __restrict__ Wres,
                                                      const float* __restrict__ agg,
                                                      const float* __restrict__ dinv,
                                                      const float* __restrict__ bconv,
                                                      const float* __restrict__ bres,
                                                      float* __restrict__ X,
                                                      float* __restrict__ Y, int N) {
    __shared__ float sB[NHID * NHID];
    for (int i = threadIdx.x; i < NHID * NHID; i += 256) sB[i] = Wres[i];
    __syncthreads();

    const int wave = threadIdx.x >> 5;
    const int lane = threadIdx.x & 31;
    const int m0 = (blockIdx.x * 8 + wave) * 16;
    if (m0 >= N) return;

    const int row  = m0 + (lane & 15);
    const int rrow = (row < N) ? row : (N - 1);
    const int koff = (lane >> 4) << 1;
    const int col  = lane & 15;

    v2f a[32];
#pragma unroll
    for (int kb = 0; kb < 32; ++kb) {
        const float* ap = h + (long long)rrow * NHID + kb * 4 + koff;
        a[kb].x = ap[0];
        a[kb].y = ap[1];
    }

    const int rbase = m0 + ((lane >> 4) << 3);
#pragma unroll
    for (int nt = 0; nt < 8; ++nt) {
        const int n0 = nt * 16;
        v8f acc = {0.f, 0.f, 0.f, 0.f, 0.f, 0.f, 0.f, 0.f};
#pragma unroll
        for (int kb = 0; kb < 32; ++kb) {
            const int k = kb * 4 + koff;
            v2f b;
            b.x = sB[k * NHID + n0 + col];
            b.y = sB[(k + 1) * NHID + n0 + col];
            acc = wmma_f32_4(a[kb], b, acc);
        }
        const float bc = bconv[n0 + col];
        const float br = bres[n0 + col];
#pragma unroll
        for (int j = 0; j < 8; ++j) {
            const int r = rbase + j;
            if (r < N) {
                const long long o = (long long)r * NHID + n0 + col;
                const float di = dinv[r];
                const float z = agg[o] + h[o] * (di * di) + bc - acc[j] - br;
                const float xo = X[o];
                const float yo = Y[o];
                const float yn = yo + DT * (tanhf(z) - ALPHA * yo - GAMMA * xo);
                Y[o] = yn;
                X[o] = xo + DT * yn;
            }
        }
    }
}

// ---------------- host-side launcher ----------------

extern "C" void kernel_launch(void* const* d_in, const int* in_sizes, int n_in,
                              void* d_out, int out_size, void* d_ws, size_t ws_size,
                              hipStream_t stream) {
    const float* x    = (const float*)d_in[0];
    const int*   ei   = (const int*)d_in[1];    // edge_index [2, E] (int32: jax x64 disabled)
    const float* W    = (const float*)d_in[2];
    const float* b    = (const float*)d_in[3];
    const float* Wres = (const float*)d_in[4];
    const float* bres = (const float*)d_in[5];

    const int N = in_sizes[0] / NHID;
    const int E = in_sizes[1] / 2;
    const int* src = ei;
    const int* dst = ei + E;

    float* X = (float*)d_out;          // output X lives in d_out
    const long long nh = (long long)N * NHID;

    float* ws   = (float*)d_ws;
    float* Y    = ws;                  // N*128
    float* h    = ws + nh;             // N*128
    float* agg  = ws + 2 * nh;         // N*128
    float* dinv = ws + 3 * nh;         // N (deg -> dinv in place)

    const int ptb = 256;
    const int nhBlocks   = (int)((nh + ptb - 1) / ptb);
    const int nBlocks    = (N + ptb - 1) / ptb;
    const int eBlocks    = (E + ptb - 1) / ptb;
    const int edgeBlocks = (int)(((long long)E * 32 + ptb - 1) / ptb);
    const int gemmBlocks = (N + 127) / 128;  // 8 waves/block * 16 rows/wave

    // X = Y = tanh(x)
    init_kernel<<<nhBlocks, ptb, 0, stream>>>(x, X, Y, nh);

    // degree normalization: dinv = rsqrt(deg + 1)
    fill_zero_kernel<<<nBlocks, ptb, 0, stream>>>(dinv, (long long)N);
    deg_count_kernel<<<eBlocks, ptb, 0, stream>>>(dst, dinv, E);
    dinv_kernel<<<nBlocks, ptb, 0, stream>>>(dinv, N);

    for (int layer = 0; layer < 4; ++layer) {
        gemm_h_kernel<<<gemmBlocks, ptb, 0, stream>>>(X, W, h, N);
        fill_zero_kernel<<<nhBlocks, ptb, 0, stream>>>(agg, nh);
        edge_kernel<<<edgeBlocks, ptb, 0, stream>>>(src, dst, h, dinv, agg, E);
        combine_kernel<<<gemmBlocks, ptb, 0, stream>>>(h, Wres, agg, dinv, b, bres, X, Y, N);
    }
}